// CCM_43757126812144
// MI455X (gfx1250) — compile-verified
//
#include <hip/hip_runtime.h>

// Problem constants (from reference): m:(8,27,1000,257) f32, x:(8,257,1000,2) f32
// out:(8,257,1000,2) f32.
constexpr int Bc = 8;
constexpr int Tc = 1000;
constexpr int Fc = 257;
constexpr int TT = 8;            // t-tile per block (1000 % 8 == 0)
constexpr int ROWS = TT + 2;     // t window rows: t0-2 .. t0+TT-1
constexpr int COLS = Fc + 2;     // f window cols: -1 .. 257
constexpr int NTHREADS = 288;    // 9 wave32s; lanes 0..256 map to f

__global__ __launch_bounds__(NTHREADS)
void ccm_conv_kernel(const float* __restrict__ m,
                     const float* __restrict__ x,
                     float* __restrict__ out)
{
    // complex x window tile: [row][col][re,im]
    __shared__ float tile[ROWS * COLS * 2];

    const int tid = threadIdx.x;
    const int t0  = blockIdx.x * TT;
    const int b   = blockIdx.y;

    // ---- Stage x window into LDS with async global->LDS (gfx1250 ASYNC path) ----
    // x[b, f, t, c] is strided by 8KB across f: per-lane async gather to LDS
    // avoids the VGPR round-trip and overlaps with the halo zero-fill.
    const unsigned long long xbase = (unsigned long long)(x + (size_t)b * Fc * Tc * 2);
    for (int idx = tid; idx < ROWS * COLS; idx += NTHREADS) {
        const int r  = idx / COLS;
        const int c  = idx - r * COLS;
        const int tx = t0 - 2 + r;     // tx < Tc always (t0+TT-1 <= 999)
        const int fx = c - 1;
        const unsigned ldsoff = (unsigned)(size_t)(&tile[idx * 2]); // LDS byte offset
        if (tx >= 0 && fx >= 0 && fx < Fc) {
            const unsigned long long ga =
                xbase + (((size_t)fx * Tc + tx) * 2ull) * sizeof(float);
            asm volatile("global_load_async_to_lds_b64 %0, %1, off"
                         :: "v"(ldsoff), "v"(ga) : "memory");
        } else {
            tile[idx * 2 + 0] = 0.0f;  // zero padding (t<0, f=-1, f=257)
            tile[idx * 2 + 1] = 0.0f;
        }
    }
    asm volatile("s_wait_asynccnt 0" ::: "memory");
    __syncthreads();

    // ---- Compute: stream m (non-temporal, read-once), reuse x from LDS ----
    if (tid < Fc) {
        const int f = tid;
        // m[b, ch, t, f]; per-(tt,k) we need ch = k, k+9, k+18
        const float* mp = m + ((size_t)b * 27 * Tc + t0) * (size_t)Fc + f;
        const float SQ = 0.8660254037844386f;  // sqrt(3)/2
        // out[b, f, t, (re,im)] : 16B-aligned at even t (t0 % 8 == 0)
        float4* op = (float4*)(out + (((size_t)b * Fc + f) * Tc + t0) * 2);

        for (int t2 = 0; t2 < TT / 2; ++t2) {
            float4 acc = make_float4(0.f, 0.f, 0.f, 0.f); // er0, ei0, er1, ei1
#pragma unroll
            for (int u = 0; u < 2; ++u) {
                const int tt = t2 * 2 + u;
                float er = 0.0f, ei = 0.0f;
#pragma unroll
                for (int k = 0; k < 9; ++k) {
                    const int i = k / 3, j = k - 3 * (k / 3);
                    const float m0 = __builtin_nontemporal_load(
                        mp + ((size_t)(k      ) * Tc + tt) * Fc);
                    const float m1 = __builtin_nontemporal_load(
                        mp + ((size_t)(k +  9) * Tc + tt) * Fc);
                    const float m2 = __builtin_nontemporal_load(
                        mp + ((size_t)(k + 18) * Tc + tt) * Fc);
                    const float hr = m0 - 0.5f * (m1 + m2);
                    const float hi = SQ * (m1 - m2);
                    // ds_load_b64: stride-2-dword lane pattern, bank-conflict-free
                    const float2 xv =
                        *(const float2*)&tile[(((tt + i) * COLS) + (f + j)) * 2];
                    er = fmaf(hr, xv.x, er);
                    er = fmaf(-hi, xv.y, er);
                    ei = fmaf(hr, xv.y, ei);
                    ei = fmaf(hi, xv.x, ei);
                }
                if (u == 0) { acc.x = er; acc.y = ei; }
                else        { acc.z = er; acc.w = ei; }
            }
            op[t2] = acc;  // b128 store: 16B contiguous per thread
        }
    }
}

extern "C" void kernel_launch(void* const* d_in, const int* in_sizes, int n_in,
                              void* d_out, int out_size, void* d_ws, size_t ws_size,
                              hipStream_t stream) {
    const float* m = (const float*)d_in[0];
    const float* x = (const float*)d_in[1];
    float* out     = (float*)d_out;
    dim3 grid(Tc / TT, Bc);   // 125 x 8 blocks
    ccm_conv_kernel<<<grid, NTHREADS, 0, stream>>>(m, x, out);
}